// ConnectivityLoss_43344809951442
// MI455X (gfx1250) — compile-verified
//
#include <hip/hip_runtime.h>
#include <hip/hip_bf16.h>
#include <math.h>
#include <stdint.h>

#define BATCH  4
#define NSEED  32
#define HD     256
#define WD     256
#define HF     1024
#define WF     1024
#define STEPS  15
#define EPSV   1e-7f

typedef unsigned u32x4 __attribute__((ext_vector_type(4)));
typedef unsigned u32x8 __attribute__((ext_vector_type(8)));

// ---------------------------------------------------------------------------
// Kernel A: softmax(C=2) -> road prob -> bilinear /4 downsample; GT stride /4
// road_ds[b,i,j] = 0.25 * sum_{r in {4i+1,4i+2}, c in {4j+1,4j+2}} sigmoid(x1-x0)
// gt_ds[b,i,j]   = (float)valid_label[b, 4i, 4j]
// ---------------------------------------------------------------------------
__global__ void cls_downsample_kernel(const float* __restrict__ cls,
                                      const int*   __restrict__ lab,
                                      float* __restrict__ road,
                                      float* __restrict__ gt)
{
    int idx = blockIdx.x * blockDim.x + threadIdx.x;
    if (idx >= BATCH * HD * WD) return;
    int j = idx & (WD - 1);
    int i = (idx >> 8) & (HD - 1);
    int b = idx >> 16;
    int r0 = 4 * i + 1, c0 = 4 * j + 1;
    const float* x0 = cls + (size_t)(b * 2 + 0) * HF * WF;
    const float* x1 = cls + (size_t)(b * 2 + 1) * HF * WF;
    float acc = 0.f;
#pragma unroll
    for (int dr = 0; dr < 2; ++dr)
#pragma unroll
        for (int dc = 0; dc < 2; ++dc) {
            size_t o = (size_t)(r0 + dr) * WF + (c0 + dc);
            float d = x1[o] - x0[o];
            acc += 1.0f / (1.0f + expf(-d));
        }
    road[idx] = 0.25f * acc;
    gt[idx]   = (float)lab[((size_t)b * HF + 4 * i) * WF + 4 * j];
}

// ---------------------------------------------------------------------------
// TDM stage of one 256x256 fp32 plane: single tensor_load_to_lds DMA.
// D# per CDNA5 ISA ch.8: group0 = {count, lds_addr, global_addr, type=2},
// group1 = {data_size=4B, tensor 256x256 stride 256, tile 256x256}.
// 2D tensor -> descriptor groups 2/3 disabled (VADDR2/3 = null).
// Issued by ONE wave (TDM ignores EXEC; per-wave issue would DMA 32x).
// ---------------------------------------------------------------------------
__device__ __forceinline__ void tdm_load_plane(const float* __restrict__ gsrc,
                                               float* sDst)
{
    unsigned           lds = (unsigned)(uintptr_t)sDst;    // LDS byte address
    unsigned long long ga  = (unsigned long long)(uintptr_t)gsrc;
    u32x4 g0;
    g0[0] = 1u;                                   // count=1, user mode
    g0[1] = lds;                                  // lds_addr
    g0[2] = (unsigned)ga;                         // global_addr[31:0]
    g0[3] = (unsigned)(ga >> 32) | (2u << 30);    // global_addr[56:32]|type=2
    u32x8 g1;
    g1[0] = 2u << 16;      // workgroup_mask=0, data_size=2 (4 bytes)
    g1[1] = 256u << 16;    // atomic_barrier_addr=0; tensor_dim0[15:0]=256
    g1[2] = 256u << 16;    // tensor_dim0[31:16]=0;  tensor_dim1[15:0]=256
    g1[3] = 256u << 16;    // tensor_dim1[31:16]=0;  tile_dim0=256
    g1[4] = 256u;          // tile_dim1=256, tile_dim2=0
    g1[5] = 256u;          // tensor_dim0_stride[31:0]=256
    g1[6] = 0u;            // tensor_dim0_stride[47:32]=0, dim1_stride lo=0
    g1[7] = 0u;            // tensor_dim1_stride hi=0
    asm volatile("tensor_load_to_lds %0, %1" :: "s"(g0), "s"(g1) : "memory");
}

// ---------------------------------------------------------------------------
// Async global -> LDS stage of one 256x256 fp32 plane (ASYNCcnt path).
// 1024 threads, each copies 64 contiguous floats (16 x b128 transfers).
// ---------------------------------------------------------------------------
__device__ __forceinline__ void async_load_plane(const float* __restrict__ gsrc,
                                                 float* sDst, int tid)
{
    // Low 32 bits of the generic pointer to a __shared__ object are the LDS
    // byte address (shared aperture base has zero low word on gfx1250).
    unsigned           lds = (unsigned)(uintptr_t)(&sDst[tid * 64]);
    unsigned long long gp  = (unsigned long long)(uintptr_t)(gsrc + tid * 64);
#pragma unroll
    for (int c = 0; c < 16; ++c) {
        asm volatile("global_load_async_to_lds_b128 %0, %1, off"
                     :
                     : "v"(lds + 16u * c), "v"(gp + 16ull * c)
                     : "memory");
    }
    asm volatile("s_wait_asynccnt 0x0" ::: "memory");
    __syncthreads();
}

// ---------------------------------------------------------------------------
// 15 iterations of s = max(seed, min(limit, maxpool7x7(s))) entirely in LDS.
// Separable pool, processed in 16-row strips, updated in place using a 3-row
// halo buffer that preserves old values across the strip boundary.
// ---------------------------------------------------------------------------
__device__ __forceinline__ void flood(const float* __restrict__ limit,
                                      const float* __restrict__ seedPlane,
                                      float* sS, float* sH, float* sHalo,
                                      int tid)
{
    for (int it = 0; it < STEPS; ++it) {
        for (int s0 = 0; s0 < HD; s0 += 16) {
            // Phase 1: horizontal 1x7 max for rows s0-3 .. s0+18 into sH.
            // Rows < s0 come from the halo (old values), rows >= s0 from sS
            // (not yet overwritten). Out-of-range rows get -inf.
#pragma unroll
            for (int p = 0; p < 6; ++p) {
                int e = tid + p * 1024;
                if (e < 22 * WD) {
                    int h = e >> 8, c = e & 255;
                    int rr = s0 - 3 + h;
                    float m = -INFINITY;
                    if (rr >= 0 && rr < HD) {
                        const float* rowp = (rr < s0)
                                                ? &sHalo[(rr - (s0 - 3)) * WD]
                                                : &sS[rr * WD];
                        int clo = (c - 3 < 0) ? 0 : c - 3;
                        int chi = (c + 3 > WD - 1) ? WD - 1 : c + 3;
                        for (int cc = clo; cc <= chi; ++cc)
                            m = fmaxf(m, rowp[cc]);
                    }
                    sH[e] = m;
                }
            }
            __syncthreads();
            // Phase 2: vertical 7x1 max over sH + min/max combine -> regs.
            // Also save the bottom 3 old rows of this strip into the halo.
            float outv[4];
#pragma unroll
            for (int q = 0; q < 4; ++q) {
                int e = q * 1024 + tid;
                int r = s0 + (e >> 8), c = e & 255;
                float m = -INFINITY;
                int hb = r - s0;  // sH row of (r-3)
#pragma unroll
                for (int dh = 0; dh < 7; ++dh)
                    m = fmaxf(m, sH[(hb + dh) * WD + c]);
                float lim = limit[r * WD + c];
                float sd  = seedPlane[r * WD + c];
                outv[q] = fmaxf(sd, fminf(lim, m));
            }
            if (tid < 3 * WD) sHalo[tid] = sS[(s0 + 13) * WD + tid];
            __syncthreads();
            // Phase 3: commit the strip.
#pragma unroll
            for (int q = 0; q < 4; ++q) {
                int e = q * 1024 + tid;
                sS[s0 * WD + e] = outv[q];
            }
            __syncthreads();
        }
    }
}

// ---------------------------------------------------------------------------
// Kernel B: one workgroup per (b, seed-k). Runs gt flood fill (binary ->
// packed to bits), reloads seeds, runs pred flood fill, accumulates BCE.
// ---------------------------------------------------------------------------
__global__ __launch_bounds__(1024) void flood_bce_kernel(
    const float* __restrict__ road, const float* __restrict__ gtds,
    const float* __restrict__ seeds, float* __restrict__ partials)
{
    __shared__ __align__(16) float sS[HD * WD];      // 256 KB plane
    __shared__ float    sH[22 * WD];                 //  22 KB hmax strip
    __shared__ float    sHalo[3 * WD];               //   3 KB old-row halo
    __shared__ unsigned sGtBits[(HD * WD) / 32];     //   8 KB binary s_gt
    __shared__ float    sRed[32];

    const int tid = threadIdx.x;
    const int b   = blockIdx.x >> 5;
    const int k   = blockIdx.x & 31;
    const float* seedPlane = seeds + (size_t)(b * NSEED + k) * HD * WD;

    // ---- pass 1: s_gt (values stay exactly in {0,1}) ----
    // Stage seeds via the Tensor Data Mover: one DMA, issued by wave 0 only.
    if (tid < 32) {
        tdm_load_plane(seedPlane, sS);
        __builtin_amdgcn_s_wait_tensorcnt(0);
    }
    __syncthreads();
    flood(gtds + (size_t)b * HD * WD, seedPlane, sS, sH, sHalo, tid);
#pragma unroll
    for (int q = 0; q < 2; ++q) {
        int w = tid * 2 + q;
        unsigned bits = 0u;
        int base = w * 32;
        for (int i = 0; i < 32; ++i)
            bits |= (sS[base + i] > 0.5f) ? (1u << i) : 0u;
        sGtBits[w] = bits;
    }
    __syncthreads();

    // ---- pass 2: s_pred (seeds re-staged via per-lane async b128 path) ----
    async_load_plane(seedPlane, sS, tid);
    flood(road + (size_t)b * HD * WD, seedPlane, sS, sH, sHalo, tid);

    // ---- BCE partial sum ----
    float acc = 0.f;
    for (int q = 0; q < 64; ++q) {
        int e = q * 1024 + tid;
        float p = sS[e];
        p = fminf(fmaxf(p, EPSV), 1.0f - EPSV);
        unsigned g = (sGtBits[e >> 5] >> (e & 31)) & 1u;
        acc += g ? (-logf(p)) : (-logf(1.0f - p));
    }
    // wave32 shuffle reduce, then cross-wave via LDS (fixed order).
#pragma unroll
    for (int o = 16; o > 0; o >>= 1) acc += __shfl_down(acc, o, 32);
    int lane = tid & 31, wave = tid >> 5;
    if (lane == 0) sRed[wave] = acc;
    __syncthreads();
    if (wave == 0) {
        float v = sRed[lane];
#pragma unroll
        for (int o = 16; o > 0; o >>= 1) v += __shfl_down(v, o, 32);
        if (lane == 0) partials[blockIdx.x] = v;
    }
}

// ---------------------------------------------------------------------------
// Kernel C: deterministic fixed-order reduction of the 128 block partials.
// ---------------------------------------------------------------------------
__global__ void reduce_kernel(const float* __restrict__ partials,
                              float* __restrict__ out)
{
    __shared__ float r[128];
    int tid = threadIdx.x;
    r[tid] = partials[tid];
    __syncthreads();
    for (int s = 64; s > 0; s >>= 1) {
        if (tid < s) r[tid] += r[tid + s];
        __syncthreads();
    }
    if (tid == 0)
        out[0] = 0.5f * r[0] / (float)(BATCH * NSEED * HD * WD);
}

// ---------------------------------------------------------------------------
extern "C" void kernel_launch(void* const* d_in, const int* in_sizes, int n_in,
                              void* d_out, int out_size, void* d_ws,
                              size_t ws_size, hipStream_t stream)
{
    (void)in_sizes; (void)n_in; (void)out_size; (void)ws_size;
    const float* cls   = (const float*)d_in[0];  // (4,2,1024,1024) f32
    const int*   lab   = (const int*)  d_in[1];  // (4,1024,1024)   i32
    const float* seeds = (const float*)d_in[2];  // (4,32,256,256)  f32

    float* ws       = (float*)d_ws;
    float* road     = ws;                    // 262144 floats
    float* gt       = ws + 262144;           // 262144 floats
    float* partials = ws + 524288;           // 128 floats

    cls_downsample_kernel<<<(BATCH * HD * WD + 255) / 256, 256, 0, stream>>>(
        cls, lab, road, gt);
    flood_bce_kernel<<<BATCH * NSEED, 1024, 0, stream>>>(road, gt, seeds,
                                                         partials);
    reduce_kernel<<<1, 128, 0, stream>>>(partials, (float*)d_out);
}